// RGBLoss_32890859552952
// MI455X (gfx1250) — compile-verified
//
#include <hip/hip_runtime.h>
#include <math.h>

// ---------------------------------------------------------------------------
// RGBLoss for MI455X (gfx1250, wave32).
//
// Key simplification: geo_coords are a permutation slice of D^3 -> unique, so
// the dense (256^3) scatter/gather in the reference is an identity:
//   surf    = geo_feats < 1.0
//   pts_rgb = rgb_feats
// Pipeline (all fp32):
//   k_init   : zbuf <- +inf bits, accumulators <- 0
//   k_setup  : per-view R = T[:3,:3] @ RotY(angle), c = T[:3,3] + offset
//   k_zmin   : per point x 4 views, project, atomicMin(zbuf) via float bits
//   k_accum  : winners (z <= zbuf) atomicAdd rgb sums + counts
//   k_loss   : per pixel masked L1 vs aux view; wave reduction done with
//              V_WMMA_F32_16X16X4_F32 (exact f32 matrix-engine reduction)
//   k_final  : loss = sum_v  (n>0 ? l / max(n,1) : 0),  n = 3*sum(mask)
// ---------------------------------------------------------------------------

#define HH     240
#define WW     320
#define PIX    (HH * WW)          // 76800
#define NVIEW  4
#define VOXf   0.03f
#define CENTf  3.825f

// workspace layout (in floats)
#define WS_ZBUF   0
#define WS_NUM    (NVIEW * PIX)                 // 307200
#define WS_CNT    (WS_NUM + NVIEW * PIX * 3)    // 1228800
#define WS_RMAT   (WS_CNT + NVIEW * PIX)        // 1536000
#define WS_CVEC   (WS_RMAT + NVIEW * 9)         // 1536036
#define WS_LACC   (WS_CVEC + NVIEW * 3)         // 1536048
#define WS_NACC   (WS_LACC + NVIEW)             // 1536052
#define WS_TOTAL  (WS_NACC + NVIEW)             // 1536056 floats (~6.1 MB)

typedef float v2f __attribute__((ext_vector_type(2)));
typedef float v8f __attribute__((ext_vector_type(8)));

#if __has_builtin(__builtin_amdgcn_wmma_f32_16x16x4_f32)
#define HAVE_WMMA_F32X4 1
#endif

__constant__ float c_offsets[NVIEW * 3] = {
    0.0f, 0.0f, 0.0f,
    0.8f, 0.0f, 0.15f,
    0.75f, 0.0f, 0.1f,
    -0.85f, 0.0f, 0.1f};
__constant__ float c_angles[NVIEW] = {0.0f, 0.0f, 0.3f, 2.15f};

// ---------------------------------------------------------------------------
__global__ void k_init(unsigned int* ws, int nzbuf, int ntotal) {
  int i = blockIdx.x * blockDim.x + threadIdx.x;
  if (i >= ntotal) return;
  ws[i] = (i < nzbuf) ? 0x7f800000u /* +inf */ : 0u;
}

// 4 threads: build per-view rotation (row-major R[i][j]) and camera center.
__global__ void k_setup(const float* __restrict__ poses,  // (1,4,4,4)
                        float* __restrict__ Rm, float* __restrict__ Cv) {
  int v = threadIdx.x;
  if (v >= NVIEW) return;
  const float* T = poses + v * 16;
  float ca = cosf(c_angles[v]);
  float sa = sinf(c_angles[v]);
  for (int i = 0; i < 3; ++i) {
    // R = T[:3,:3] @ [[ca,0,sa],[0,1,0],[-sa,0,ca]]
    Rm[v * 9 + i * 3 + 0] = T[i * 4 + 0] * ca - T[i * 4 + 2] * sa;
    Rm[v * 9 + i * 3 + 1] = T[i * 4 + 1];
    Rm[v * 9 + i * 3 + 2] = T[i * 4 + 0] * sa + T[i * 4 + 2] * ca;
    Cv[v * 3 + i] = T[i * 4 + 3] + c_offsets[v * 3 + i];
  }
}

// ---------------------------------------------------------------------------
// Pass 1: z-buffer min via uint atomicMin (positive floats order == bit order)
__global__ void k_zmin(const int* __restrict__ coords,
                       const float* __restrict__ gfeat,
                       const float* __restrict__ Rm, const float* __restrict__ Cv,
                       unsigned int* __restrict__ zbuf, int N) {
  int i = blockIdx.x * blockDim.x + threadIdx.x;
  if (i >= N) return;
  if (!(gfeat[i] < 1.0f)) return;  // surf mask
  float px = (float)coords[3 * i + 0] * VOXf - CENTf;
  float py = (float)coords[3 * i + 1] * VOXf - CENTf;
  float pz = (float)coords[3 * i + 2] * VOXf - CENTf;
#pragma unroll
  for (int v = 0; v < NVIEW; ++v) {
    const float* r = Rm + v * 9;
    const float* c = Cv + v * 3;
    float d0 = px - c[0], d1 = py - c[1], d2 = pz - c[2];
    float z = d0 * r[2] + d1 * r[5] + d2 * r[8];
    if (!(z > 1e-4f)) continue;
    float x = d0 * r[0] + d1 * r[3] + d2 * r[6];
    float y = d0 * r[1] + d1 * r[4] + d2 * r[7];
    float uf = rintf(240.0f * x / z + 160.0f);  // RNE == jnp.round
    float vf = rintf(240.0f * y / z + 120.0f);
    if (uf < 0.0f || uf > (float)(WW - 1) || vf < 0.0f || vf > (float)(HH - 1)) continue;
    int u = (int)uf, vv = (int)vf;
    atomicMin(zbuf + v * PIX + vv * WW + u, __float_as_uint(z));
  }
}

// Pass 2: winners accumulate rgb sums and counts (L2-resident f32 atomics)
__global__ void k_accum(const int* __restrict__ coords,
                        const float* __restrict__ gfeat,
                        const float* __restrict__ rgb,
                        const float* __restrict__ Rm, const float* __restrict__ Cv,
                        const unsigned int* __restrict__ zbuf,
                        float* __restrict__ num, float* __restrict__ cnt, int N) {
  int i = blockIdx.x * blockDim.x + threadIdx.x;
  if (i >= N) return;
  if (!(gfeat[i] < 1.0f)) return;
  float px = (float)coords[3 * i + 0] * VOXf - CENTf;
  float py = (float)coords[3 * i + 1] * VOXf - CENTf;
  float pz = (float)coords[3 * i + 2] * VOXf - CENTf;
  float r0 = rgb[3 * i + 0], r1 = rgb[3 * i + 1], r2 = rgb[3 * i + 2];
#pragma unroll
  for (int v = 0; v < NVIEW; ++v) {
    const float* r = Rm + v * 9;
    const float* c = Cv + v * 3;
    float d0 = px - c[0], d1 = py - c[1], d2 = pz - c[2];
    float z = d0 * r[2] + d1 * r[5] + d2 * r[8];
    if (!(z > 1e-4f)) continue;
    float x = d0 * r[0] + d1 * r[3] + d2 * r[6];
    float y = d0 * r[1] + d1 * r[4] + d2 * r[7];
    float uf = rintf(240.0f * x / z + 160.0f);
    float vf = rintf(240.0f * y / z + 120.0f);
    if (uf < 0.0f || uf > (float)(WW - 1) || vf < 0.0f || vf > (float)(HH - 1)) continue;
    int u = (int)uf, vv = (int)vf;
    int pix = v * PIX + vv * WW + u;
    float zb = __uint_as_float(zbuf[pix]);
    if (z <= zb) {
      atomicAdd(num + pix * 3 + 0, r0);
      atomicAdd(num + pix * 3 + 1, r1);
      atomicAdd(num + pix * 3 + 2, r2);
      atomicAdd(cnt + pix, 1.0f);
    }
  }
}

// ---------------------------------------------------------------------------
// Exact f32 wave-wide reduction of (l, n) pairs on the matrix engine.
// A(16x4): lane L contributes a=[l,n] -> A[L%16][{0,1}] (lanes 0-15) or
// A[L%16][{2,3}] (lanes 16-31). Selector B routes K=0,1 (lanes 0-15) to columns
// 0-7 and K=2,3 (lanes 16-31) to columns 8-15. D[m][n] then holds pairwise row
// sums; adding the 8 accumulator VGPRs + one shfl_xor(16) gives per-half-wave sums of l+n.
__device__ __forceinline__ void wave_reduce_add(float lval, float nval,
                                                float* laccv, float* naccv) {
  unsigned lane = (unsigned)(threadIdx.x & 31);
#if defined(HAVE_WMMA_F32X4)
  unsigned n16 = lane & 15u;
  float sel = (((lane < 16u) == (n16 < 8u)) ? 1.0f : 0.0f);
  v2f a;
  a[0] = lval;
  a[1] = nval;
  v2f b;
  b[0] = sel;
  b[1] = sel;
  v8f c = {0.f, 0.f, 0.f, 0.f, 0.f, 0.f, 0.f, 0.f};
  c = __builtin_amdgcn_wmma_f32_16x16x4_f32(false, a, false, b, (short)0, c,
                                            false, false);
  float s = c[0] + c[1] + c[2] + c[3] + c[4] + c[5] + c[6] + c[7];
  s += __shfl_xor(s, 16, 32);
  if (lane == 0u) atomicAdd(laccv, s);   // col 0 -> l+n summed over lanes 0-15
  if (lane == 8u) atomicAdd(naccv, s);   // col 8 -> l+n summed over lanes 16-31
#else
  float L = lval, Nn = nval;
  for (int off = 16; off > 0; off >>= 1) {
    L += __shfl_xor(L, off, 32);
    Nn += __shfl_xor(Nn, off, 32);
  }
  if (lane == 0u) {
    atomicAdd(laccv, L);
    atomicAdd(naccv, Nn);
  }
#endif
}

// Pass 3: masked L1 per pixel; grid = (PIX/256, NVIEW). Exactly one pixel per
// thread, so EXEC is full at the WMMA (required by the ISA).
__global__ void k_loss(const float* __restrict__ num, const float* __restrict__ cnt,
                       const float* __restrict__ aux,  // (4,3,H,W)
                       float* __restrict__ lacc, float* __restrict__ nacc) {
  int v = blockIdx.y;
  float l = 0.0f, nm = 0.0f;
  int p = blockIdx.x * blockDim.x + threadIdx.x;
  if (p < PIX) {
    float cn = cnt[v * PIX + p];
    float den = fmaxf(cn, 1.0f);
    const float* nb = num + (v * PIX + p) * 3;
#pragma unroll
    for (int ch = 0; ch < 3; ++ch) {
      float img = nb[ch] / den;
      if (img != 0.0f) {
        float vw = aux[(v * 3 + ch) * PIX + p];
        l += fabsf(img - vw);
        nm += 1.0f;
      }
    }
  }
  wave_reduce_add(l, nm, &lacc[v], &nacc[v]);
}

__global__ void k_final(const float* __restrict__ lacc,
                        const float* __restrict__ nacc, float* __restrict__ out) {
  if (threadIdx.x == 0 && blockIdx.x == 0) {
    float s = 0.0f;
    for (int v = 0; v < NVIEW; ++v) {
      float n = 3.0f * nacc[v];
      if (n > 0.0f) s += lacc[v] / fmaxf(n, 1.0f);
    }
    out[0] = s;
  }
}

// ---------------------------------------------------------------------------
extern "C" void kernel_launch(void* const* d_in, const int* in_sizes, int n_in,
                              void* d_out, int out_size, void* d_ws, size_t ws_size,
                              hipStream_t stream) {
  const int* coords = (const int*)d_in[0];    // (N,3) int32
  const float* gfeat = (const float*)d_in[1]; // (N,)
  const float* rgb = (const float*)d_in[2];   // (N,3)
  const float* aux = (const float*)d_in[3];   // (1,4,3,H,W)
  const float* poses = (const float*)d_in[4]; // (1,4,4,4)
  int N = in_sizes[1];

  float* ws = (float*)d_ws;  // requires ~6.15 MB of scratch
  unsigned int* zbuf = (unsigned int*)(ws + WS_ZBUF);
  float* num = ws + WS_NUM;
  float* cnt = ws + WS_CNT;
  float* Rm = ws + WS_RMAT;
  float* Cv = ws + WS_CVEC;
  float* lacc = ws + WS_LACC;
  float* nacc = ws + WS_NACC;

  k_init<<<(WS_TOTAL + 255) / 256, 256, 0, stream>>>((unsigned int*)ws,
                                                     NVIEW * PIX, WS_TOTAL);
  k_setup<<<1, 4, 0, stream>>>(poses, Rm, Cv);
  k_zmin<<<(N + 255) / 256, 256, 0, stream>>>(coords, gfeat, Rm, Cv, zbuf, N);
  k_accum<<<(N + 255) / 256, 256, 0, stream>>>(coords, gfeat, rgb, Rm, Cv, zbuf,
                                               num, cnt, N);
  dim3 lgrid(PIX / 256, NVIEW);
  k_loss<<<lgrid, 256, 0, stream>>>(num, cnt, aux, lacc, nacc);
  k_final<<<1, 1, 0, stream>>>(lacc, nacc, (float*)d_out);
}